// EdgeGCNModel_81046032875707
// MI455X (gfx1250) — compile-verified
//
#include <hip/hip_runtime.h>
#include <hip/hip_bf16.h>
#include <math.h>

// ---------------- model dimensions ----------------
#define BGR   64          // graphs
#define NNODE 512         // nodes per graph
#define NTOT  32768       // BGR*NNODE
#define HD    128         // hidden
#define IC4V  2048        // rb2 width
#define KN1   410         // pool1 keep
#define KN2   328         // pool2 keep
#define KP    416         // KN1 padded to multiple of 16
#define EDGES 524288      // BGR*8192
#define ZDIM  4352        // (HD+IC4V)*2
#define EPSV  1e-5f

static_assert(EDGES % 16 == 0, "edge tiling");
static_assert(KP % 16 == 0 && KP % 4 == 0, "pad");

typedef __attribute__((ext_vector_type(2))) float v2f;
typedef __attribute__((ext_vector_type(8))) float v8f;

__device__ __forceinline__ v8f v8f_zero() {
  v8f v;
#pragma unroll
  for (int i = 0; i < 8; ++i) v[i] = 0.f;
  return v;
}

// CDNA5 fp32 WMMA: D(16x16,f32) = A(16x4,f32) * B(4x16,f32) + C
__device__ __forceinline__ v8f wmma4(v2f a, v2f b, v8f c) {
  return __builtin_amdgcn_wmma_f32_16x16x4_f32(false, a, false, b, (short)0, c,
                                               false, false);
}

// =====================================================================
// Generic WMMA GEMM: C[M,N] = A[M,K] @ B[K,N] (+bias)(+relu)
// One wave computes a 16-row x (NTILES*16)-col tile. NTILES is a template
// parameter so every accumulator access is a compile-time index -> the
// acc array lives in VGPRs (no scratch spills around v_wmma).
// Batched operands via rowsPerBatch + strides (rowsPerBatch % 16 == 0).
// =====================================================================
template <int NTILES>
__global__ void gemm_wmma(const float* __restrict__ A, const float* __restrict__ Bw,
                          const float* __restrict__ bias, float* __restrict__ C,
                          int n0base, int K, int lda, int ldb, int ldc,
                          int rowsPerBatch, long sA, long sB, long sC, int relu) {
  const int lane  = threadIdx.x;          // 0..31 (wave32)
  const int tile  = blockIdx.x;           // row tile
  const int n0    = n0base + blockIdx.y * 128;
  const int batch = (tile * 16) / rowsPerBatch;
  const int rbase = tile * 16 - batch * rowsPerBatch;
  const int mlane = lane & 15;
  const int kh    = (lane >> 4) * 2;      // K-pair base for this half-wave
  const float* Arow = A + batch * sA + (long)(rbase + mlane) * lda;
  const float* Bb   = Bw + batch * sB + n0 + mlane;
  float*       Cb   = C  + batch * sC;

  v8f acc[NTILES];
#pragma unroll
  for (int t = 0; t < NTILES; ++t) acc[t] = v8f_zero();

  for (int kk = 0; kk < K; kk += 4) {
    v2f a = *(const v2f*)(Arow + kk + kh);          // global_load_b64
    const float* Bk = Bb + (long)(kk + kh) * ldb;
#pragma unroll
    for (int t = 0; t < NTILES; ++t) {
      v2f b;
      b.x = Bk[t * 16];
      b.y = Bk[t * 16 + ldb];
      acc[t] = wmma4(a, b, acc[t]);
    }
  }
  const int mh = (lane >> 4) * 8;
#pragma unroll
  for (int t = 0; t < NTILES; ++t) {
    int ncol = n0 + t * 16 + mlane;
    float bv = bias ? bias[ncol] : 0.f;
#pragma unroll
    for (int r = 0; r < 8; ++r) {
      float v = acc[t][r] + bv;
      if (relu) v = fmaxf(v, 0.f);
      Cb[(long)(rbase + r + mh) * ldc + ncol] = v;
    }
  }
}

// =====================================================================
// EdgeConv: out[dst] = max over edges of relu(W @ [x_i, x_j - x_i] + b)
// One wave per 16 edges; feature tile staged in LDS; WMMA over K=2*CIN;
// scatter via uint atomicMax (values are >=0 after relu; accumulator
// pre-zeroed so empty segments -> 0, matching the reference fill).
// =====================================================================
template <int CIN>
__global__ void edgeconv_wmma(const float* __restrict__ X, const int* __restrict__ srcI,
                              const int* __restrict__ dstI, const float* __restrict__ W,
                              const float* __restrict__ bias, unsigned* __restrict__ Out) {
  constexpr int KK  = 2 * CIN;
  constexpr int STR = (CIN == 128) ? 260 : 8;  // bank-conflict-free stride (even)
  __shared__ float feat[16 * STR];
  __shared__ int dloc[16];
  const int lane = threadIdx.x;
  const long e0 = (long)blockIdx.x * 16;
  if (lane < 16) dloc[lane] = dstI[e0 + lane];
  if constexpr (CIN == 128) {
    for (int m = 0; m < 16; ++m) {
      int s = srcI[e0 + m], d = dstI[e0 + m];
      const float4* xi = (const float4*)(X + (long)d * 128);
      const float4* xj = (const float4*)(X + (long)s * 128);
      float4 a = xi[lane], b = xj[lane];
      float* fm = feat + m * STR;
      int c = lane * 4;
      fm[c + 0] = a.x; fm[c + 1] = a.y; fm[c + 2] = a.z; fm[c + 3] = a.w;
      fm[128 + c + 0] = b.x - a.x; fm[128 + c + 1] = b.y - a.y;
      fm[128 + c + 2] = b.z - a.z; fm[128 + c + 3] = b.w - a.w;
    }
  } else {
    if (lane < 16) {
      int s = srcI[e0 + lane], d = dstI[e0 + lane];
      float xi0 = X[(long)d * 2], xi1 = X[(long)d * 2 + 1];
      float xj0 = X[(long)s * 2], xj1 = X[(long)s * 2 + 1];
      float* fm = feat + lane * STR;
      fm[0] = xi0; fm[1] = xi1; fm[2] = xj0 - xi0; fm[3] = xj1 - xi1;
    }
  }
  __syncthreads();
  const int mlane = lane & 15;
  const int kh = (lane >> 4) * 2;
  v8f acc[8];
#pragma unroll
  for (int t = 0; t < 8; ++t) acc[t] = v8f_zero();
  for (int kk = 0; kk < KK; kk += 4) {
    v2f a = *(const v2f*)(feat + mlane * STR + kk + kh);  // ds_load_b64
#pragma unroll
    for (int t = 0; t < 8; ++t) {
      const float* Wk = W + (long)(kk + kh) * HD + t * 16 + mlane;
      v2f b; b.x = Wk[0]; b.y = Wk[HD];
      acc[t] = wmma4(a, b, acc[t]);
    }
  }
  const int mh = (lane >> 4) * 8;
#pragma unroll
  for (int t = 0; t < 8; ++t) {
    int ncol = t * 16 + mlane;
    float bv = bias[ncol];
#pragma unroll
    for (int r = 0; r < 8; ++r) {
      float v = fmaxf(acc[t][r] + bv, 0.f);
      atomicMax(Out + (long)dloc[r + mh] * HD + ncol, __float_as_uint(v));
    }
  }
}

// ---------------- elementwise / reduction helpers ----------------
__global__ void fill_kernel(float* p, float v, long n) {
  long i = (long)blockIdx.x * blockDim.x + threadIdx.x;
  if (i < n) p[i] = v;
}

__global__ void deg_acc(const int* __restrict__ dstI, float* deg, int E) {
  int e = blockIdx.x * blockDim.x + threadIdx.x;
  if (e < E) atomicAdd(deg + dstI[e], 1.0f);
}

__global__ void rsqrt_ip(float* p, long n) {
  long i = (long)blockIdx.x * blockDim.x + threadIdx.x;
  if (i < n) p[i] = rsqrtf(p[i]);
}

// y[r,c] = x[r,0]*W[0,c] + x[r,1]*W[1,c] (+b)   (K=2 linear, 2->128)
__global__ void lin2(const float* __restrict__ X2, const float* __restrict__ W,
                     const float* __restrict__ bias, float* __restrict__ Y, int M) {
  long idx = (long)blockIdx.x * blockDim.x + threadIdx.x;
  if (idx >= (long)M * HD) return;
  int c = idx & (HD - 1);
  long r = idx >> 7;
  float v = X2[r * 2] * W[c] + X2[r * 2 + 1] * W[HD + c];
  if (bias) v += bias[c];
  Y[idx] = v;
}

// y[r,0..1] = h[r,:] @ W[128,2] + b
__global__ void lin_out2(const float* __restrict__ Hin, const float* __restrict__ W,
                         const float* __restrict__ b2, float* __restrict__ Y, int M) {
  int r = blockIdx.x * blockDim.x + threadIdx.x;
  if (r >= M) return;
  float a0 = b2[0], a1 = b2[1];
  const float* h = Hin + (long)r * HD;
  for (int k = 0; k < HD; ++k) { a0 += h[k] * W[2 * k]; a1 += h[k] * W[2 * k + 1]; }
  Y[r * 2] = a0; Y[r * 2 + 1] = a1;
}

// agg[dst,c] += xw[src,c]*di[src]*di[dst]   (one block per edge, 128 ch)
__global__ void gcn_agg(const float* __restrict__ XW, const int* __restrict__ srcI,
                        const int* __restrict__ dstI, const float* __restrict__ di,
                        float* __restrict__ agg) {
  int e = blockIdx.x, c = threadIdx.x;
  int s = srcI[e], d = dstI[e];
  float coef = di[s] * di[d];
  atomicAdd(agg + (long)d * HD + c, XW[(long)s * HD + c] * coef);
}

// agg += xw*di^2 + bias  (self-loop term)
__global__ void gcn_combine(float* __restrict__ agg, const float* __restrict__ XW,
                            const float* __restrict__ di, const float* __restrict__ bias,
                            int M) {
  long idx = (long)blockIdx.x * blockDim.x + threadIdx.x;
  if (idx >= (long)M * HD) return;
  int c = idx & (HD - 1);
  long r = idx >> 7;
  float dd = di[r];
  agg[idx] = agg[idx] + XW[idx] * dd * dd + bias[c];
}

// per-channel batch stats -> mean, rstd=rsqrt(var+eps); biased variance
__global__ void bn_stats(const float* __restrict__ Z, float* __restrict__ meanO,
                         float* __restrict__ rstdO, int ld, int batches,
                         int validRows, int rowStride) {
  int c = blockIdx.x;
  __shared__ float ss[256], ss2[256];
  float s = 0.f, s2 = 0.f;
  for (int b = 0; b < batches; ++b) {
    const float* base = Z + (long)b * rowStride * ld + c;
    for (int r = threadIdx.x; r < validRows; r += blockDim.x) {
      float v = base[(long)r * ld];
      s += v; s2 += v * v;
    }
  }
  ss[threadIdx.x] = s; ss2[threadIdx.x] = s2;
  __syncthreads();
  for (int w = 128; w > 0; w >>= 1) {
    if (threadIdx.x < w) { ss[threadIdx.x] += ss[threadIdx.x + w]; ss2[threadIdx.x] += ss2[threadIdx.x + w]; }
    __syncthreads();
  }
  if (threadIdx.x == 0) {
    float inv = 1.0f / ((float)batches * (float)validRows);
    float m = ss[0] * inv;
    float var = ss2[0] * inv - m * m;
    meanO[c] = m;
    rstdO[c] = rsqrtf(var + EPSV);
  }
}

// y = (z-mean)*rstd*g + be (+resid)(+relu); padded rows -> 0
__global__ void bn_apply(const float* __restrict__ Z, const float* __restrict__ mean,
                         const float* __restrict__ rstd, const float* __restrict__ g,
                         const float* __restrict__ be, const float* __restrict__ resid,
                         float* __restrict__ Y, int ld, int batches, int validRows,
                         int rowStride, int relu) {
  long idx = (long)blockIdx.x * blockDim.x + threadIdx.x;
  long total = (long)batches * rowStride * ld;
  if (idx >= total) return;
  int c = (int)(idx % ld);
  long row = idx / ld;
  int b = (int)(row / rowStride);
  int r = (int)(row - (long)b * rowStride);
  float y = 0.f;
  if (r < validRows) {
    float v = (Z[idx] - mean[c]) * rstd[c] * g[c] + be[c];
    if (resid) v += resid[idx];
    if (relu) v = fmaxf(v, 0.f);
    y = v;
  }
  Y[idx] = y;
}

// =====================================================================
// TopK pooling: scores=sigmoid(h.attn/||attn||); bitonic sort 512 in LDS
// =====================================================================
__global__ void topk_pool(const float* __restrict__ Hin, const float* __restrict__ attn,
                          int C, int n, int k, int ldh, int graphStride,
                          int* __restrict__ perm, float* __restrict__ val) {
  int b = blockIdx.x, tid = threadIdx.x;   // 512 threads
  __shared__ float red[512];
  __shared__ float sc[512];
  __shared__ int si[512];
  float an = 0.f;
  for (int c = tid; c < C; c += 512) { float a = attn[c]; an += a * a; }
  red[tid] = an;
  __syncthreads();
  for (int w = 256; w > 0; w >>= 1) {
    if (tid < w) red[tid] += red[tid + w];
    __syncthreads();
  }
  float invn = rsqrtf(red[0]);
  __syncthreads();
  float s = -INFINITY;
  if (tid < n) {
    const float* row = Hin + ((long)b * graphStride + tid) * ldh;
    float dot = 0.f;
    for (int c = 0; c < C; ++c) dot += row[c] * attn[c];
    s = 1.0f / (1.0f + expf(-dot * invn));
  }
  sc[tid] = s; si[tid] = tid;
  __syncthreads();
  for (int kb = 2; kb <= 512; kb <<= 1) {
    for (int j = kb >> 1; j > 0; j >>= 1) {
      int ixj = tid ^ j;
      if (ixj > tid) {
        bool descRegion = ((tid & kb) == 0);
        float s1 = sc[tid], s2 = sc[ixj];
        int i1 = si[tid], i2 = si[ixj];
        bool aFirst = (s1 > s2) || (s1 == s2 && i1 < i2);  // desc order
        bool doSwap = descRegion ? !aFirst : aFirst;
        if (doSwap) { sc[tid] = s2; sc[ixj] = s1; si[tid] = i2; si[ixj] = i1; }
      }
      __syncthreads();
    }
  }
  if (tid < k) { perm[b * k + tid] = si[tid]; val[b * k + tid] = sc[tid]; }
}

// hp[b,r,:] = H[b,perm[r],:]*val[r]; pad rows zero
__global__ void gather_scale(const float* __restrict__ Hin, const int* __restrict__ perm,
                             const float* __restrict__ val, float* __restrict__ Y,
                             int C, int k, int padRows, int graphStride, int ldh) {
  int rowIdx = blockIdx.x;
  int b = rowIdx / padRows, r = rowIdx - b * padRows;
  float* out = Y + (long)rowIdx * C;
  if (r >= k) { for (int c = threadIdx.x; c < C; c += blockDim.x) out[c] = 0.f; return; }
  int sr = perm[b * k + r];
  float v = val[b * k + r];
  const float* srcp = Hin + ((long)b * graphStride + sr) * ldh;
  for (int c = threadIdx.x; c < C; c += blockDim.x) out[c] = srcp[c] * v;
}

// z[b, zoff+c]=mean_r(H[b,perm[r],c]*val[r]); z[b, zoff+C+c]=max_r(...)
__global__ void meanmax_perm(const float* __restrict__ Hin, const int* __restrict__ perm,
                             const float* __restrict__ val, float* __restrict__ Z,
                             int ldz, int zoff, int C, int k, int graphStride, int ldh) {
  int b = blockIdx.x;
  int c = blockIdx.y * blockDim.x + threadIdx.x;
  if (c >= C) return;
  float s = 0.f, mx = -INFINITY;
  for (int r = 0; r < k; ++r) {
    float v = Hin[((long)b * graphStride + perm[b * k + r]) * ldh + c] * val[b * k + r];
    s += v; mx = fmaxf(mx, v);
  }
  Z[(long)b * ldz + zoff + c] = s / (float)k;
  Z[(long)b * ldz + zoff + C + c] = mx;
}

// ---------------- dense adjacency ----------------
__global__ void a_scatter(const int* __restrict__ srcI, const int* __restrict__ dstI,
                          const float* __restrict__ w, float* __restrict__ A, int E) {
  int e = blockIdx.x * blockDim.x + threadIdx.x;
  if (e >= E) return;
  int s = srcI[e], d = dstI[e];
  int b = s >> 9;                // / NNODE
  atomicAdd(A + (long)b * NNODE * NNODE + (long)(d - (b << 9)) * NNODE + (s - (b << 9)), w[e]);
}

// As[b,i,j] = (i,j<K1)? A[b,perm[i],perm[j]] + (i==j) : 0
__global__ void as_build(const float* __restrict__ A, const int* __restrict__ perm,
                         float* __restrict__ As) {
  long idx = (long)blockIdx.x * blockDim.x + threadIdx.x;
  if (idx >= (long)BGR * KP * KP) return;
  int j = (int)(idx % KP);
  long t = idx / KP;
  int i = (int)(t % KP);
  int b = (int)(t / KP);
  float v = 0.f;
  if (i < KN1 && j < KN1) {
    int pi = perm[b * KN1 + i], pj = perm[b * KN1 + j];
    v = A[(long)b * NNODE * NNODE + (long)pi * NNODE + pj];
    if (i == j) v += 1.f;
  }
  As[idx] = v;
}

// di[b,i] = rsqrt(sum_j Ah[i,j]),  Ah = A2 with diag forced to 1; pad->0
__global__ void an_rowsum(const float* __restrict__ A2, float* __restrict__ di) {
  int rowIdx = blockIdx.x;
  int b = rowIdx / KP, i = rowIdx - b * KP;
  (void)b;
  __shared__ float red[128];
  float s = 0.f;
  if (i < KN1) {
    const float* row = A2 + (long)rowIdx * KP;
    for (int j = threadIdx.x; j < KN1; j += blockDim.x)
      s += (j == i) ? 1.f : row[j];
  }
  red[threadIdx.x] = s;
  __syncthreads();
  for (int w = 64; w > 0; w >>= 1) {
    if (threadIdx.x < w) red[threadIdx.x] += red[threadIdx.x + w];
    __syncthreads();
  }
  if (threadIdx.x == 0) di[rowIdx] = (i < KN1) ? rsqrtf(red[0]) : 0.f;
}

// An[i,j] = di[i]*Ah[i,j]*di[j] in place
__global__ void an_apply(float* __restrict__ A2, const float* __restrict__ di) {
  long idx = (long)blockIdx.x * blockDim.x + threadIdx.x;
  if (idx >= (long)BGR * KP * KP) return;
  int j = (int)(idx % KP);
  long t = idx / KP;
  int i = (int)(t % KP);
  int b = (int)(t / KP);
  float ah = (i < KN1 && j < KN1) ? ((i == j) ? 1.f : A2[idx]) : 0.f;
  A2[idx] = di[b * KP + i] * ah * di[b * KP + j];
}

// fc3 + log_softmax -> d_out [64,2]
__global__ void head_kernel(const float* __restrict__ Zin, const float* __restrict__ W,
                            const float* __restrict__ b2, float* __restrict__ outp) {
  int r = threadIdx.x;
  if (r >= BGR) return;
  float a0 = b2[0], a1 = b2[1];
  const float* h = Zin + (long)r * HD;
  for (int k = 0; k < HD; ++k) { a0 += h[k] * W[2 * k]; a1 += h[k] * W[2 * k + 1]; }
  float mx = fmaxf(a0, a1);
  float lse = mx + logf(expf(a0 - mx) + expf(a1 - mx));
  outp[r * 2] = a0 - lse;
  outp[r * 2 + 1] = a1 - lse;
}

// ---------------- host side ----------------
static inline void launch_gemm(hipStream_t st, const float* A, const float* B,
                               const float* bias, float* C, int M, int N, int K,
                               int lda, int ldb, int ldc, int rowsPerBatch,
                               long sA, long sB, long sC, int relu) {
  const dim3 blk(32);
  const int mt = M / 16;
  const int nfull = N / 128;
  const int rem = (N - nfull * 128) / 16;
  if (nfull > 0)
    hipLaunchKernelGGL((gemm_wmma<8>), dim3(mt, nfull), blk, 0, st, A, B, bias, C,
                       0, K, lda, ldb, ldc, rowsPerBatch, sA, sB, sC, relu);
  if (rem > 0) {
    const int n0 = nfull * 128;
    switch (rem) {
      case 1: hipLaunchKernelGGL((gemm_wmma<1>), dim3(mt, 1), blk, 0, st, A, B, bias, C, n0, K, lda, ldb, ldc, rowsPerBatch, sA, sB, sC, relu); break;
      case 2: hipLaunchKernelGGL((gemm_wmma<2>), dim3(mt, 1), blk, 0, st, A, B, bias, C, n0, K, lda, ldb, ldc, rowsPerBatch, sA, sB, sC, relu); break;
      case 3: hipLaunchKernelGGL((gemm_wmma<3>), dim3(mt, 1), blk, 0, st, A, B, bias, C, n0, K, lda, ldb, ldc, rowsPerBatch, sA, sB, sC, relu); break;
      case 4: hipLaunchKernelGGL((gemm_wmma<4>), dim3(mt, 1), blk, 0, st, A, B, bias, C, n0, K, lda, ldb, ldc, rowsPerBatch, sA, sB, sC, relu); break;
      case 5: hipLaunchKernelGGL((gemm_wmma<5>), dim3(mt, 1), blk, 0, st, A, B, bias, C, n0, K, lda, ldb, ldc, rowsPerBatch, sA, sB, sC, relu); break;
      case 6: hipLaunchKernelGGL((gemm_wmma<6>), dim3(mt, 1), blk, 0, st, A, B, bias, C, n0, K, lda, ldb, ldc, rowsPerBatch, sA, sB, sC, relu); break;
      default: hipLaunchKernelGGL((gemm_wmma<7>), dim3(mt, 1), blk, 0, st, A, B, bias, C, n0, K, lda, ldb, ldc, rowsPerBatch, sA, sB, sC, relu); break;
    }
  }
}

// input tensor order: setup_inputs() dict insertion order, nested dicts
// flattened in insertion order.
enum {
  IN_X = 0, IN_EIDX, IN_EW,
  IN_MLP1_W, IN_MLP1_B, IN_MLP2_W, IN_MLP2_B, IN_MLP3_W, IN_MLP3_B,
  IN_FC4_W, IN_FC4_B, IN_FC5_W, IN_FC5_B, IN_FC6_W, IN_FC6_B,
  IN_BN3_G, IN_BN3_B, IN_BN4_G, IN_BN4_B, IN_BN5_G, IN_BN5_B,
  IN_LIN_W, IN_LIN_B,
  IN_RB1_C1_W, IN_RB1_C1_B, IN_RB1_C2_W, IN_RB1_C2_B,
  IN_RB1_BN1_G, IN_RB1_BN1_B, IN_RB1_BN2_G, IN_RB1_BN2_B,
  IN_RB1_RES_W, IN_RB1_RES_B,
  IN_P1ATTN,
  IN_RB2_C1_W, IN_RB2_C1_B, IN_RB2_C2_W, IN_RB2_C2_B,
  IN_RB2_BN1_G, IN_RB2_BN1_B, IN_RB2_BN2_G, IN_RB2_BN2_B,
  IN_RB2_RES_W, IN_RB2_RES_B,
  IN_P2ATTN,
  IN_FC1_W, IN_FC1_B, IN_BN1_G, IN_BN1_B,
  IN_FC2_W, IN_FC2_B, IN_BN2_G, IN_BN2_B,
  IN_FC3_W, IN_FC3_B
};

extern "C" void kernel_launch(void* const* d_in, const int* in_sizes, int n_in,
                              void* d_out, int out_size, void* d_ws, size_t ws_size,
                              hipStream_t stream) {
  (void)in_sizes; (void)n_in; (void)out_size; (void)ws_size;
  const float* x    = (const float*)d_in[IN_X];
  const int* eidx   = (const int*)d_in[IN_EIDX];
  const int* srcI   = eidx;
  const int* dstI   = eidx + EDGES;
  const float* ew   = (const float*)d_in[IN_EW];
  auto F = [&](int i) { return (const float*)d_in[i]; };

  // ---- workspace layout (floats) ----
  float* W0 = (float*)d_ws;
  size_t o = 0;
  auto alloc = [&](size_t n) { float* p = W0 + o; o += (n + 63) & ~(size_t)63; return p; };
  float* bufA  = alloc((size_t)NTOT * HD);
  float* bufB  = alloc((size_t)NTOT * HD);
  float* bufC  = alloc((size_t)NTOT * HD);
  float* x2f   = alloc((size_t)NTOT * 2);
  float* deg   = alloc(NTOT);
  float* meanv = alloc(IC4V);
  float* rstdv = alloc(IC4V);
  float* val1  = alloc((size_t)BGR * KN1);
  int*   perm1 = (int*)alloc((size_t)BGR * KN1);
  float* val2  = alloc((size_t)BGR * KN2);
  int*   perm2 = (int*)alloc((size_t)BGR * KN2);
  float* di2   = alloc((size_t)BGR * KP);
  float* zfeat = alloc((size_t)BGR * ZDIM);
  float* zb1   = alloc((size_t)BGR * HD);
  float* zb2   = alloc((size_t)BGR * HD);
  float* hp    = alloc((size_t)BGR * KP * HD);
  float* Afull = alloc((size_t)BGR * NNODE * NNODE);   // reused for A2 / An
  float* Aspd  = alloc((size_t)BGR * KP * KP);
  float* P     = alloc((size_t)BGR * KP * IC4V);
  float* Q     = alloc((size_t)BGR * KP * IC4V);

  const size_t nhBytes = (size_t)NTOT * HD * sizeof(float);
  const int ETB = EDGES / 16;
  const int MD = BGR * KP;        // dense phase rows (26624)
  const long sAA = (long)KP * KP;
  const long sAX = (long)KP * IC4V;

  // ===== EdgeConv stack =====
  hipMemsetAsync(bufA, 0, nhBytes, stream);
  hipLaunchKernelGGL((edgeconv_wmma<2>), dim3(ETB), dim3(32), 0, stream,
                     x, srcI, dstI, F(IN_MLP1_W), F(IN_MLP1_B), (unsigned*)bufA);
  launch_gemm(stream, bufA, F(IN_FC4_W), F(IN_FC4_B), bufB, NTOT, HD, HD, HD, HD, HD, NTOT, 0, 0, 0, 1);
  hipLaunchKernelGGL(bn_stats, dim3(HD), dim3(256), 0, stream, bufB, meanv, rstdv, HD, 1, NTOT, NTOT);
  hipLaunchKernelGGL(bn_apply, dim3((NTOT * HD + 255) / 256), dim3(256), 0, stream,
                     bufB, meanv, rstdv, F(IN_BN3_G), F(IN_BN3_B), (const float*)nullptr,
                     bufB, HD, 1, NTOT, NTOT, 0);

  hipMemsetAsync(bufA, 0, nhBytes, stream);
  hipLaunchKernelGGL((edgeconv_wmma<128>), dim3(ETB), dim3(32), 0, stream,
                     bufB, srcI, dstI, F(IN_MLP2_W), F(IN_MLP2_B), (unsigned*)bufA);
  launch_gemm(stream, bufA, F(IN_FC5_W), F(IN_FC5_B), bufB, NTOT, HD, HD, HD, HD, HD, NTOT, 0, 0, 0, 1);
  hipLaunchKernelGGL(bn_stats, dim3(HD), dim3(256), 0, stream, bufB, meanv, rstdv, HD, 1, NTOT, NTOT);
  hipLaunchKernelGGL(bn_apply, dim3((NTOT * HD + 255) / 256), dim3(256), 0, stream,
                     bufB, meanv, rstdv, F(IN_BN4_G), F(IN_BN4_B), (const float*)nullptr,
                     bufB, HD, 1, NTOT, NTOT, 0);

  hipMemsetAsync(bufA, 0, nhBytes, stream);
  hipLaunchKernelGGL((edgeconv_wmma<128>), dim3(ETB), dim3(32), 0, stream,
                     bufB, srcI, dstI, F(IN_MLP3_W), F(IN_MLP3_B), (unsigned*)bufA);
  launch_gemm(stream, bufA, F(IN_FC6_W), F(IN_FC6_B), bufB, NTOT, HD, HD, HD, HD, HD, NTOT, 0, 0, 0, 1);
  hipLaunchKernelGGL(bn_stats, dim3(HD), dim3(256), 0, stream, bufB, meanv, rstdv, HD, 1, NTOT, NTOT);
  hipLaunchKernelGGL(bn_apply, dim3((NTOT * HD + 255) / 256), dim3(256), 0, stream,
                     bufB, meanv, rstdv, F(IN_BN5_G), F(IN_BN5_B), (const float*)nullptr,
                     bufB, HD, 1, NTOT, NTOT, 0);

  // lin: H -> 2
  hipLaunchKernelGGL(lin_out2, dim3(NTOT / 128), dim3(128), 0, stream,
                     bufB, F(IN_LIN_W), F(IN_LIN_B), x2f, NTOT);

  // ===== rb1 (sparse GCN residual block) =====
  hipLaunchKernelGGL(fill_kernel, dim3((NTOT + 255) / 256), dim3(256), 0, stream, deg, 1.0f, (long)NTOT);
  hipLaunchKernelGGL(deg_acc, dim3(EDGES / 256), dim3(256), 0, stream, dstI, deg, EDGES);
  hipLaunchKernelGGL(rsqrt_ip, dim3((NTOT + 255) / 256), dim3(256), 0, stream, deg, (long)NTOT);

  hipLaunchKernelGGL(lin2, dim3((NTOT * HD + 255) / 256), dim3(256), 0, stream,
                     x2f, F(IN_RB1_C1_W), (const float*)nullptr, bufA, NTOT);
  hipMemsetAsync(bufC, 0, nhBytes, stream);
  hipLaunchKernelGGL(gcn_agg, dim3(EDGES), dim3(HD), 0, stream, bufA, srcI, dstI, deg, bufC);
  hipLaunchKernelGGL(gcn_combine, dim3((NTOT * HD + 255) / 256), dim3(256), 0, stream,
                     bufC, bufA, deg, F(IN_RB1_C1_B), NTOT);
  hipLaunchKernelGGL(bn_stats, dim3(HD), dim3(256), 0, stream, bufC, meanv, rstdv, HD, 1, NTOT, NTOT);
  hipLaunchKernelGGL(bn_apply, dim3((NTOT * HD + 255) / 256), dim3(256), 0, stream,
                     bufC, meanv, rstdv, F(IN_RB1_BN1_G), F(IN_RB1_BN1_B), (const float*)nullptr,
                     bufC, HD, 1, NTOT, NTOT, 1);

  launch_gemm(stream, bufC, F(IN_RB1_C2_W), nullptr, bufA, NTOT, HD, HD, HD, HD, HD, NTOT, 0, 0, 0, 0);
  hipMemsetAsync(bufB, 0, nhBytes, stream);
  hipLaunchKernelGGL(gcn_agg, dim3(EDGES), dim3(HD), 0, stream, bufA, srcI, dstI, deg, bufB);
  hipLaunchKernelGGL(gcn_combine, dim3((NTOT * HD + 255) / 256), dim3(256), 0, stream,
                     bufB, bufA, deg, F(IN_RB1_C2_B), NTOT);
  hipLaunchKernelGGL(bn_stats, dim3(HD), dim3(256), 0, stream, bufB, meanv, rstdv, HD, 1, NTOT, NTOT);
  hipLaunchKernelGGL(lin2, dim3((NTOT * HD + 255) / 256), dim3(256), 0, stream,
                     x2f, F(IN_RB1_RES_W), F(IN_RB1_RES_B), bufA, NTOT);
  hipLaunchKernelGGL(bn_apply, dim3((NTOT * HD + 255) / 256), dim3(256), 0, stream,
                     bufB, meanv, rstdv, F(IN_RB1_BN2_G), F(IN_RB1_BN2_B), bufA,
                     bufB, HD, 1, NTOT, NTOT, 1);          // h in bufB

  // ===== pool1 =====
  hipLaunchKernelGGL(topk_pool, dim3(BGR), dim3(512), 0, stream,
                     bufB, F(IN_P1ATTN), HD, NNODE, KN1, HD, NNODE, perm1, val1);
  hipLaunchKernelGGL(gather_scale, dim3(BGR * KP), dim3(128), 0, stream,
                     bufB, perm1, val1, hp, HD, KN1, KP, NNODE, HD);
  hipLaunchKernelGGL(meanmax_perm, dim3(BGR, 1), dim3(128), 0, stream,
                     bufB, perm1, val1, zfeat, ZDIM, 0, HD, KN1, NNODE, HD);

  // ===== dense adjacency: A -> As -> A2 -> An =====
  hipMemsetAsync(Afull, 0, (size_t)BGR * NNODE * NNODE * sizeof(float), stream);
  hipLaunchKernelGGL(a_scatter, dim3(EDGES / 256), dim3(256), 0, stream, srcI, dstI, ew, Afull, EDGES);
  hipLaunchKernelGGL(as_build, dim3((int)(((long)BGR * KP * KP + 255) / 256)), dim3(256), 0, stream,
                     Afull, perm1, Aspd);
  launch_gemm(stream, Aspd, Aspd, nullptr, Afull, MD, KP, KP, KP, KP, KP, KP, sAA, sAA, sAA, 0);
  hipLaunchKernelGGL(an_rowsum, dim3(BGR * KP), dim3(128), 0, stream, Afull, di2);
  hipLaunchKernelGGL(an_apply, dim3((int)(((long)BGR * KP * KP + 255) / 256)), dim3(256), 0, stream,
                     Afull, di2);                          // Afull now holds An

  // ===== rb2 (dense GCN residual block) =====
  const long denseElems = (long)MD * IC4V;
  launch_gemm(stream, hp, F(IN_RB2_C1_W), nullptr, P, MD, IC4V, HD, HD, IC4V, IC4V, MD, 0, 0, 0, 0);
  launch_gemm(stream, Afull, P, F(IN_RB2_C1_B), Q, MD, IC4V, KP, KP, IC4V, IC4V, KP, sAA, sAX, sAX, 0);
  hipLaunchKernelGGL(bn_stats, dim3(IC4V), dim3(256), 0, stream, Q, meanv, rstdv, IC4V, BGR, KN1, KP);
  hipLaunchKernelGGL(bn_apply, dim3((int)((denseElems + 255) / 256)), dim3(256), 0, stream,
                     Q, meanv, rstdv, F(IN_RB2_BN1_G), F(IN_RB2_BN1_B), (const float*)nullptr,
                     P, IC4V, BGR, KN1, KP, 1);            // h in P (pads zeroed)
  launch_gemm(stream, P, F(IN_RB2_C2_W), nullptr, Q, MD, IC4V, IC4V, IC4V, IC4V, IC4V, MD, 0, 0, 0, 0);
  launch_gemm(stream, Afull, Q, F(IN_RB2_C2_B), P, MD, IC4V, KP, KP, IC4V, IC4V, KP, sAA, sAX, sAX, 0);
  launch_gemm(stream, hp, F(IN_RB2_RES_W), F(IN_RB2_RES_B), Q, MD, IC4V, HD, HD, IC4V, IC4V, MD, 0, 0, 0, 0);
  hipLaunchKernelGGL(bn_stats, dim3(IC4V), dim3(256), 0, stream, P, meanv, rstdv, IC4V, BGR, KN1, KP);
  hipLaunchKernelGGL(bn_apply, dim3((int)((denseElems + 255) / 256)), dim3(256), 0, stream,
                     P, meanv, rstdv, F(IN_RB2_BN2_G), F(IN_RB2_BN2_B), Q,
                     Q, IC4V, BGR, KN1, KP, 1);            // h2 in Q

  // ===== pool2 + readout =====
  hipLaunchKernelGGL(topk_pool, dim3(BGR), dim3(512), 0, stream,
                     Q, F(IN_P2ATTN), IC4V, KN1, KN2, IC4V, KP, perm2, val2);
  hipLaunchKernelGGL(meanmax_perm, dim3(BGR, IC4V / 128), dim3(128), 0, stream,
                     Q, perm2, val2, zfeat, ZDIM, 2 * HD, IC4V, KN2, KP, IC4V);

  // ===== classifier head =====
  launch_gemm(stream, zfeat, F(IN_FC1_W), F(IN_FC1_B), zb1, BGR, HD, ZDIM, ZDIM, HD, HD, BGR, 0, 0, 0, 1);
  hipLaunchKernelGGL(bn_stats, dim3(HD), dim3(256), 0, stream, zb1, meanv, rstdv, HD, 1, BGR, BGR);
  hipLaunchKernelGGL(bn_apply, dim3((BGR * HD + 255) / 256), dim3(256), 0, stream,
                     zb1, meanv, rstdv, F(IN_BN1_G), F(IN_BN1_B), (const float*)nullptr,
                     zb1, HD, 1, BGR, BGR, 0);
  launch_gemm(stream, zb1, F(IN_FC2_W), F(IN_FC2_B), zb2, BGR, HD, HD, HD, HD, HD, BGR, 0, 0, 0, 1);
  hipLaunchKernelGGL(bn_stats, dim3(HD), dim3(256), 0, stream, zb2, meanv, rstdv, HD, 1, BGR, BGR);
  hipLaunchKernelGGL(bn_apply, dim3((BGR * HD + 255) / 256), dim3(256), 0, stream,
                     zb2, meanv, rstdv, F(IN_BN2_G), F(IN_BN2_B), (const float*)nullptr,
                     zb2, HD, 1, BGR, BGR, 0);
  hipLaunchKernelGGL(head_kernel, dim3(1), dim3(64), 0, stream,
                     zb2, F(IN_FC3_W), F(IN_FC3_B), (float*)d_out);
}